// GradientFeatureExtractor_18322330485157
// MI455X (gfx1250) — compile-verified
//
#include <hip/hip_runtime.h>

// GradientFeatureExtractor for MI455X (gfx1250, wave32).
// Memory-bound two-pass reduction; WMMA f32 16x16x4 (B=ones) used as the
// in-register accumulation engine for the grad-magnitude sum.

typedef __attribute__((ext_vector_type(2))) float v2f;
typedef __attribute__((ext_vector_type(8))) float v8f;

#define TILE_ROWS 32
#define NPIX_F    1048576.0f   // 1024*1024

// Workspace layout in 32-bit words:
#define WS_SUM   0     // [64]   f32  sum(grad_mag)           (atomicAdd f32)
#define WS_MAX   64    // [64]   u32  bits of max(grad_mag)   (atomicMax u32; mags > 0)
#define WS_BSUM  128   // [64*4] f32  per-angle-bin sum(mag)
#define WS_BCNT  384   // [64*4] u32  per-angle-bin count
#define WS_CGT   640   // [64]   u32  count(mag > mean)
#define WS_WORDS 704

__global__ __launch_bounds__(256) void gfe_init(unsigned* __restrict__ ws) {
  unsigned i = blockIdx.x * 256u + threadIdx.x;
  if (i < WS_WORDS) ws[i] = 0u;
}

// Angle-bin classification (quadrants of atan2(gy, gx+1e-8), half-open bins):
//   bin0: [-pi,-pi/2)  gy<0 && gxe<0
//   bin1: [-pi/2,0)    gy<0 && gxe>=0
//   bin2: [0,pi/2)     (gy>0 && gxe>0) || (gy==0 && gxe>=0)
//   bin3: [pi/2,pi)    gy>0 && gxe<=0
//   gy==0 && gxe<0 -> angle==pi -> no bin (matches reference half-open masks)

__global__ __launch_bounds__(256) void gfe_pass1(const float* __restrict__ x,
                                                 float* __restrict__ wsf) {
  unsigned* wsu = reinterpret_cast<unsigned*>(wsf);
  const int b  = blockIdx.y;
  const int r0 = blockIdx.x * TILE_ROWS;
  const int c  = threadIdx.x * 4;           // 256 threads * 4 = full 1024-wide row
  // Branch-free left neighbor: thread 0 loads row[0] == cur.x -> gx[0] == 0.
  const int loff = (c == 0) ? 0 : (c - 1);
  const float* img = x + ((size_t)b << 20); // 1024*1024 per image

  v8f acc = {};                 // WMMA C/D accumulator (16x16 f32)
  v2f ones; ones.x = 1.0f; ones.y = 1.0f;   // B = all-ones 4x16
  float vmax = 0.0f;
  float bs0 = 0.f, bs1 = 0.f, bs2 = 0.f, bs3 = 0.f;
  unsigned c0 = 0, c1 = 0, c2 = 0, c3 = 0;  // wave-uniform (ballot-based)

  // Row above the tile; for the first tile use row 0 itself -> grad_y == 0 there.
  const int rp = (r0 == 0) ? 0 : (r0 - 1);
  float4 prev = *reinterpret_cast<const float4*>(img + ((size_t)rp << 10) + c);

#pragma unroll 4
  for (int r = r0; r < r0 + TILE_ROWS; ++r) {
    const float* row = img + ((size_t)r << 10);
    float4 cur = *reinterpret_cast<const float4*>(row + c);
    float left = row[loff];             // independent load; L1 hit (same lines)

    float gx[4], gy[4], mag[4];
    gx[0] = cur.x - left;   gx[1] = cur.y - cur.x;
    gx[2] = cur.z - cur.y;  gx[3] = cur.w - cur.z;
    gy[0] = cur.x - prev.x; gy[1] = cur.y - prev.y;
    gy[2] = cur.z - prev.z; gy[3] = cur.w - prev.w;

#pragma unroll
    for (int i = 0; i < 4; ++i) {
      float t = fmaf(gy[i], gy[i], fmaf(gx[i], gx[i], 1e-8f));
      float m = __builtin_amdgcn_sqrtf(t);   // raw v_sqrt_f32 (TRANS, co-executes)
      mag[i] = m;
      vmax = fmaxf(vmax, m);
      float gxe = gx[i] + 1e-8f;
      bool gyp = gy[i] > 0.0f;
      bool gyn = gy[i] < 0.0f;
      bool gxp = gxe   > 0.0f;
      bool gxn = gxe   < 0.0f;
      bool i0 = gyn && gxn;
      bool i1 = gyn && !gxn;
      bool i3 = gyp && !gxp;
      bool i2 = (gyp && gxp) || (!gyp && !gyn && !gxn);
      bs0 += i0 ? m : 0.0f;
      bs1 += i1 ? m : 0.0f;
      bs2 += i2 ? m : 0.0f;
      bs3 += i3 ? m : 0.0f;
      // counts ride the scalar pipe: v_cmp -> s_bcnt1 -> s_add (co-issues)
      c0 += __builtin_popcount(__builtin_amdgcn_ballot_w32(i0));
      c1 += __builtin_popcount(__builtin_amdgcn_ballot_w32(i1));
      c2 += __builtin_popcount(__builtin_amdgcn_ballot_w32(i2));
      c3 += __builtin_popcount(__builtin_amdgcn_ballot_w32(i3));
    }

    // D = A * ones + C : every (m,k) slot of A holds a unique partial sum,
    // so the D matrix accumulates 128 mags per instruction across the wave.
    v2f a; a.x = mag[0] + mag[1]; a.y = mag[2] + mag[3];
    acc = __builtin_amdgcn_wmma_f32_16x16x4_f32(false, a, false, ones,
                                                (short)0, acc, false, false);
    prev = cur;
  }

  // Collapse WMMA accumulator: sum over all lanes/regs = 16 * true_sum
  // (16 identical columns), so divide by 16 after the lane reduction.
  float cs = acc[0] + acc[1] + acc[2] + acc[3] + acc[4] + acc[5] + acc[6] + acc[7];
#pragma unroll
  for (int off = 16; off >= 1; off >>= 1) {
    cs  += __shfl_xor(cs,  off, 32);
    bs0 += __shfl_xor(bs0, off, 32);
    bs1 += __shfl_xor(bs1, off, 32);
    bs2 += __shfl_xor(bs2, off, 32);
    bs3 += __shfl_xor(bs3, off, 32);
    vmax = fmaxf(vmax, __shfl_xor(vmax, off, 32));
  }
  if ((threadIdx.x & 31) == 0) {
    atomicAdd(&wsf[WS_SUM + b], cs * (1.0f / 16.0f));
    atomicMax(&wsu[WS_MAX + b], __float_as_uint(vmax));
    atomicAdd(&wsf[WS_BSUM + b * 4 + 0], bs0);
    atomicAdd(&wsf[WS_BSUM + b * 4 + 1], bs1);
    atomicAdd(&wsf[WS_BSUM + b * 4 + 2], bs2);
    atomicAdd(&wsf[WS_BSUM + b * 4 + 3], bs3);
    atomicAdd(&wsu[WS_BCNT + b * 4 + 0], c0);
    atomicAdd(&wsu[WS_BCNT + b * 4 + 1], c1);
    atomicAdd(&wsu[WS_BCNT + b * 4 + 2], c2);
    atomicAdd(&wsu[WS_BCNT + b * 4 + 3], c3);
  }
}

__global__ __launch_bounds__(256) void gfe_pass2(const float* __restrict__ x,
                                                 float* __restrict__ wsf) {
  unsigned* wsu = reinterpret_cast<unsigned*>(wsf);
  const int b  = blockIdx.y;
  const int r0 = blockIdx.x * TILE_ROWS;
  const int c  = threadIdx.x * 4;
  const int loff = (c == 0) ? 0 : (c - 1);
  const float* img = x + ((size_t)b << 20);

  const float mean = wsf[WS_SUM + b] * (1.0f / NPIX_F);
  const float thr  = mean * mean;   // mag > mean  <=>  mag^2 > mean^2 (both >= 0)

  unsigned cnt = 0;
  const int rp = (r0 == 0) ? 0 : (r0 - 1);
  float4 prev = *reinterpret_cast<const float4*>(img + ((size_t)rp << 10) + c);

#pragma unroll 4
  for (int r = r0; r < r0 + TILE_ROWS; ++r) {
    const float* row = img + ((size_t)r << 10);
    float4 cur = *reinterpret_cast<const float4*>(row + c);
    float left = row[loff];

    float gx[4], gy[4];
    gx[0] = cur.x - left;   gx[1] = cur.y - cur.x;
    gx[2] = cur.z - cur.y;  gx[3] = cur.w - cur.z;
    gy[0] = cur.x - prev.x; gy[1] = cur.y - prev.y;
    gy[2] = cur.z - prev.z; gy[3] = cur.w - prev.w;

#pragma unroll
    for (int i = 0; i < 4; ++i) {
      float t = fmaf(gy[i], gy[i], fmaf(gx[i], gx[i], 1e-8f));
      cnt += __builtin_popcount(__builtin_amdgcn_ballot_w32(t > thr));
    }
    prev = cur;
  }
  if ((threadIdx.x & 31) == 0) atomicAdd(&wsu[WS_CGT + b], cnt);
}

__global__ __launch_bounds__(64) void gfe_finalize(const float* __restrict__ wsf,
                                                   float* __restrict__ out) {
  const unsigned* wsu = reinterpret_cast<const unsigned*>(wsf);
  int b = threadIdx.x;
  if (b < 64) {
    const float inv = 1.0f / NPIX_F;
    out[b * 7 + 0] = (float)wsu[WS_CGT + b] * inv;           // edge_density
    out[b * 7 + 1] = wsf[WS_SUM + b] * inv;                  // edge_strength
    out[b * 7 + 2] = __uint_as_float(wsu[WS_MAX + b]);       // edge_max
#pragma unroll
    for (int k = 0; k < 4; ++k) {
      float num = wsf[WS_BSUM + b * 4 + k];
      float den = (float)wsu[WS_BCNT + b * 4 + k] + 1e-8f;
      out[b * 7 + 3 + k] = num / den;                        // angle features
    }
  }
}

extern "C" void kernel_launch(void* const* d_in, const int* in_sizes, int n_in,
                              void* d_out, int out_size, void* d_ws, size_t ws_size,
                              hipStream_t stream) {
  const float* x = (const float*)d_in[0];
  float* out = (float*)d_out;
  float* ws  = (float*)d_ws;

  hipLaunchKernelGGL(gfe_init, dim3((WS_WORDS + 255) / 256), dim3(256), 0, stream,
                     (unsigned*)d_ws);

  dim3 grid(1024 / TILE_ROWS, 64);   // 32 row-tiles x 64 batches = 2048 blocks
  hipLaunchKernelGGL(gfe_pass1, grid, dim3(256), 0, stream, x, ws);
  hipLaunchKernelGGL(gfe_pass2, grid, dim3(256), 0, stream, x, ws);
  hipLaunchKernelGGL(gfe_finalize, dim3(1), dim3(64), 0, stream, ws, out);
}